// SubnetworkBlock_85753317032594
// MI455X (gfx1250) — compile-verified
//
#include <hip/hip_runtime.h>
#include <math.h>

#define S_NUM  64
#define H1     10
#define H2     6
#define GRID_N 101
#define BATCH  131072

// ---- CDNA5 vector types for WMMA ----------------------------------------
typedef __attribute__((ext_vector_type(2))) float v2f;   // 16x4 / 4x16 f32 operand = 2 VGPRs
typedef __attribute__((ext_vector_type(8))) float v8f;   // 16x16 f32 C/D      = 8 VGPRs

union V8 { v8f v; float f[8]; };

// ---- hardware tanh (CDNA5 VALU transcendental v_tanh_f32) ---------------
#if __has_builtin(__builtin_amdgcn_tanhf)
__device__ __forceinline__ float fast_tanh(float x) { return __builtin_amdgcn_tanhf(x); }
#elif __has_builtin(__builtin_amdgcn_tanh_f32)
__device__ __forceinline__ float fast_tanh(float x) { return __builtin_amdgcn_tanh_f32(x); }
#else
__device__ __forceinline__ float fast_tanh(float x) {
  float r;
  asm volatile("v_tanh_f32 %0, %1\n\tv_nop" : "=v"(r) : "v"(x));
  return r;
}
#endif

// =========================================================================
// Kernel A: per-subnet grid statistics (mean, 1/max(std,EPS)) -> d_ws[128]
// =========================================================================
__global__ __launch_bounds__(S_NUM) void grid_stats_kernel(
    const float* __restrict__ W1, const float* __restrict__ b1,
    const float* __restrict__ W2, const float* __restrict__ b2,
    const float* __restrict__ W3, const float* __restrict__ b3,
    float* __restrict__ stats)
{
  const int s = threadIdx.x;           // one thread per subnet
  float w1[H1], bb1[H1], w2[H1 * H2], bb2[H2], w3[H2];
#pragma unroll
  for (int k = 0; k < H1; ++k) { w1[k] = W1[s * H1 + k]; bb1[k] = b1[s * H1 + k]; }
#pragma unroll
  for (int i = 0; i < H1 * H2; ++i) w2[i] = W2[s * H1 * H2 + i];
#pragma unroll
  for (int j = 0; j < H2; ++j) { bb2[j] = b2[s * H2 + j]; w3[j] = W3[s * H2 + j]; }
  const float bb3 = b3[s];

  float sum = 0.0f, sumsq = 0.0f;
  for (int g = 0; g < GRID_N; ++g) {
    float x = fmaf((float)g, 2.0f / (GRID_N - 1), -1.0f);   // linspace(-1,1,101)
    float h[H1];
#pragma unroll
    for (int k = 0; k < H1; ++k) h[k] = fast_tanh(fmaf(x, w1[k], bb1[k]));
    float o = bb3;
#pragma unroll
    for (int j = 0; j < H2; ++j) {
      float a = bb2[j];
#pragma unroll
      for (int i = 0; i < H1; ++i) a = fmaf(h[i], w2[i * H2 + j], a);
      o = fmaf(fast_tanh(a), w3[j], o);
    }
    sum += o;
    sumsq = fmaf(o, o, sumsq);
  }
  const float mean = sum * (1.0f / GRID_N);
  float var = fmaxf(sumsq * (1.0f / GRID_N) - mean * mean, 0.0f);
  float sd  = fmaxf(sqrtf(var), 1e-10f);
  stats[s]         = mean;
  stats[S_NUM + s] = 1.0f / sd;
}

// =========================================================================
// Kernel B: main batch pass.  One wave = one subnet, 16-row batch tiles.
//   Z(6x16) = W2^T(6x10) x H^T(10x16) + b2   via 3x V_WMMA_F32_16X16X4_F32
// D layout: VGPR r = h2-channel, lane = batch row -> layer-3 is a per-lane
// FMA chain over D VGPRs (no cross-lane ops).  Normalization folded in.
// x-loads are software-pipelined one tile ahead.
// =========================================================================
#define CHUNKS          128          // 64*128 = 8192 waves
#define WAVES_PER_BLOCK 4
#define TILE_STRIDE     (16 * S_NUM) // floats per 16-row tile

__device__ __forceinline__ void process_tile(
    float x, const v2f A[3], const float w1k[6], const float b1k[6],
    v8f cinit, const float w3i[H2], float fin, float* outp, bool lo)
{
  // B = H^T: lane n holds h[k] for its two K slots per chunk; pads -> tanh(0)=0
  v2f B0, B1, B2;
  B0.x = fast_tanh(fmaf(x, w1k[0], b1k[0]));
  B0.y = fast_tanh(fmaf(x, w1k[1], b1k[1]));
  B1.x = fast_tanh(fmaf(x, w1k[2], b1k[2]));
  B1.y = fast_tanh(fmaf(x, w1k[3], b1k[3]));
  B2.x = fast_tanh(fmaf(x, w1k[4], b1k[4]));
  B2.y = fast_tanh(fmaf(x, w1k[5], b1k[5]));

  V8 acc;
  acc.v = __builtin_amdgcn_wmma_f32_16x16x4_f32(false, A[0], false, B0, (short)0, cinit, false, false);
  acc.v = __builtin_amdgcn_wmma_f32_16x16x4_f32(false, A[1], false, B1, (short)0, acc.v, false, false);
  acc.v = __builtin_amdgcn_wmma_f32_16x16x4_f32(false, A[2], false, B2, (short)0, acc.v, false, false);

  float o = fin;
#pragma unroll
  for (int r = 0; r < H2; ++r) o = fmaf(fast_tanh(acc.f[r]), w3i[r], o);

  if (lo) *outp = o;                  // EXEC restored before next tile's WMMA
}

__global__ __launch_bounds__(WAVES_PER_BLOCK * 32) void mlp_wmma_kernel(
    const float* __restrict__ in,  const int* __restrict__ idx,
    const float* __restrict__ W1,  const float* __restrict__ b1,
    const float* __restrict__ W2,  const float* __restrict__ b2,
    const float* __restrict__ W3,  const float* __restrict__ b3,
    const float* __restrict__ stats, float* __restrict__ out)
{
  const int lane  = threadIdx.x & 31;
  const int wave  = blockIdx.x * WAVES_PER_BLOCK + (threadIdx.x >> 5);
  const int s     = wave & (S_NUM - 1);
  const int chunk = wave >> 6;
  const int rows  = BATCH / CHUNKS;       // 1024 rows per wave
  const int tiles = rows / 16;            // 64 tiles per wave
  const int b0    = chunk * rows;

  const int m    = lane & 15;             // A: M row (h2 chan) / B,D: N col (batch row)
  const int koff = (lane >= 16) ? 2 : 0;  // high half-wave carries K+2,K+3 of each chunk
  const bool lo  = (lane < 16);

  // ---- per-wave constants (amortized over 64 tiles) ----
  float w1k[6], b1k[6];
  v2f   A[3];                             // A = W2^T, K chunks 0..3 / 4..7 / 8..11 (pad 0)
#pragma unroll
  for (int c = 0; c < 3; ++c) {
    const int ka = 4 * c + koff, kb = ka + 1;
    w1k[2 * c]     = (ka < H1) ? W1[s * H1 + ka] : 0.0f;
    b1k[2 * c]     = (ka < H1) ? b1[s * H1 + ka] : 0.0f;
    w1k[2 * c + 1] = (kb < H1) ? W1[s * H1 + kb] : 0.0f;
    b1k[2 * c + 1] = (kb < H1) ? b1[s * H1 + kb] : 0.0f;
    A[c].x = (m < H2 && ka < H1) ? W2[s * H1 * H2 + ka * H2 + m] : 0.0f;
    A[c].y = (m < H2 && kb < H1) ? W2[s * H1 * H2 + kb * H2 + m] : 0.0f;
  }

  // C = b2 broadcast: VGPR r, lanes 0-15 -> M=r (b2[r], r<6); lanes 16-31 -> 0
  V8 cini;
#pragma unroll
  for (int r = 0; r < 8; ++r) cini.f[r] = 0.0f;
  if (lo) {
#pragma unroll
    for (int r = 0; r < H2; ++r) cini.f[r] = b2[s * H2 + r];
  }

  const float mean = stats[s];
  const float inv  = stats[S_NUM + s];
  float w3i[H2];
#pragma unroll
  for (int r = 0; r < H2; ++r) w3i[r] = W3[s * H2 + r] * inv;
  const float fin = (b3[s] - mean) * inv;
  const int   col = idx[s];

  const float* pin  = in  + (size_t)(b0 + m) * S_NUM + col;
  float*       pout = out + (size_t)(b0 + m) * S_NUM + s;

  // ---- software-pipelined tile loop: next x-load in flight during compute ----
  float x = *pin;
#pragma unroll 1
  for (int t = 0; t < tiles - 1; ++t) {
    pin += TILE_STRIDE;
    const float xn = *pin;                 // prefetch tile t+1 (one load in flight)
    process_tile(x, A, w1k, b1k, cini.v, w3i, fin, pout, lo);
    pout += TILE_STRIDE;
    x = xn;
  }
  process_tile(x, A, w1k, b1k, cini.v, w3i, fin, pout, lo);   // epilogue
}

// =========================================================================
extern "C" void kernel_launch(void* const* d_in, const int* in_sizes, int n_in,
                              void* d_out, int out_size, void* d_ws, size_t ws_size,
                              hipStream_t stream) {
  (void)in_sizes; (void)n_in; (void)out_size; (void)ws_size;
  const float* in  = (const float*)d_in[0];
  const int*   idx = (const int*)  d_in[1];
  const float* W1  = (const float*)d_in[2];
  const float* b1  = (const float*)d_in[3];
  const float* W2  = (const float*)d_in[4];
  const float* b2  = (const float*)d_in[5];
  const float* W3  = (const float*)d_in[6];
  const float* b3  = (const float*)d_in[7];
  float* out   = (float*)d_out;
  float* stats = (float*)d_ws;   // [0..63]=mean, [64..127]=1/max(std,eps)

  grid_stats_kernel<<<1, S_NUM, 0, stream>>>(W1, b1, W2, b2, W3, b3, stats);

  const int blocks = (S_NUM * CHUNKS) / WAVES_PER_BLOCK;   // 2048
  mlp_wmma_kernel<<<blocks, WAVES_PER_BLOCK * 32, 0, stream>>>(
      in, idx, W1, b1, W2, b2, W3, b3, stats, out);
}